// SwitchGate_12919261626593
// MI455X (gfx1250) — compile-verified
//
#include <hip/hip_runtime.h>
#include <hip/hip_bf16.h>

typedef __attribute__((ext_vector_type(2))) float v2f;
typedef __attribute__((ext_vector_type(8))) float v8f;

// ---------------------------------------------------------------------------
// Kernel 1: per-wave 16-token x 16-expert logit tile via V_WMMA_F32_16X16X4_F32,
// then softmax + top-8 mask, storing masked gate scores to out[t*16 + e].
//
// A fragment (16x4 f32): lanes 0-15 hold row M=lane, K=0..1; lanes 16-31 hold
// row M=lane-16, K=2..3  -> each lane does one b64 load of X[row][k+2h .. +1].
// B fragment (4x16 f32): lane n holds column n; VGPR0=K{0|2}, VGPR1=K{1|3}
// per half -> same b64 pattern from W[n][k+2h].
// C (16x16 f32): VGPR i, lanes 0-15 -> token i, lanes 16-31 -> token 8+i.
// ---------------------------------------------------------------------------
__global__ __launch_bounds__(256)
void switchgate_score_kernel(const float* __restrict__ X,
                             const float* __restrict__ W,
                             const float* __restrict__ bias,
                             float* __restrict__ out,
                             int D) {
    const int lane = threadIdx.x & 31;
    const int n    = lane & 15;      // expert / column index
    const int h    = lane >> 4;      // half-wave selector
    const int wave = threadIdx.x >> 5;
    const int tile = blockIdx.x * 8 + wave;   // 16 tokens per tile
    const long rowBase = (long)tile * 16;

    const float* __restrict__ xrow = X + (rowBase + n) * (long)D + 2 * h;
    const float* __restrict__ wrow = W + (long)n * D + 2 * h;

    v8f c = {0.f, 0.f, 0.f, 0.f, 0.f, 0.f, 0.f, 0.f};

    for (int k = 0; k < D; k += 16) {
#pragma unroll
        for (int u = 0; u < 4; ++u) {
            v2f a  = *(const v2f*)(xrow + k + 4 * u);
            v2f bb = *(const v2f*)(wrow + k + 4 * u);
            // D = A(16x4) * B(4x16) + C ; fp32 in / fp32 accumulate
            c = __builtin_amdgcn_wmma_f32_16x16x4_f32(
                    /*neg_a=*/false, a, /*neg_b=*/false, bb,
                    /*c_mod=*/(short)0, c, /*reuse_a=*/false, /*reuse_b=*/false);
        }
    }

    const float bias_n = bias[n];

    float sc[8];
#pragma unroll
    for (int i = 0; i < 8; ++i) sc[i] = c[i] + bias_n;

    // Softmax over the 16 experts (spread across the 16 lanes of each half),
    // then top-8 selection via rank (ties broken by lower expert index, matching
    // jax.lax.top_k stability). Masked gate score or 0 is kept.
#pragma unroll
    for (int i = 0; i < 8; ++i) {
        float s = sc[i];

        float m = s;
#pragma unroll
        for (int mask = 1; mask < 16; mask <<= 1)
            m = fmaxf(m, __shfl_xor(m, mask, 32));

        float e = __expf(s - m);
        float sum = e;
#pragma unroll
        for (int mask = 1; mask < 16; mask <<= 1)
            sum += __shfl_xor(sum, mask, 32);

        float gate = e / sum;

        int rank = 0;
#pragma unroll
        for (int j = 0; j < 16; ++j) {
            float sj = __shfl(s, j, 16);   // broadcast within 16-lane group
            rank += (sj > s) || (sj == s && j < n);
        }
        sc[i] = (rank < 8) ? gate : 0.0f;
    }

    // Store masked scores: token t = rowBase + h*8 + i, expert n.
#pragma unroll
    for (int i = 0; i < 8; ++i) {
        long t = rowBase + h * 8 + i;
        out[t * 16 + n] = sc[i];
    }
}

// ---------------------------------------------------------------------------
// Kernel 2: cross-batch normalization. denom[s,e] = sum_b masked[b,s,e] + eps;
// out = masked / denom * capacity (capacity = int(1.0*B) = 8). In-place, each
// thread exclusively owns its 8 (b,s,e) slots.
// ---------------------------------------------------------------------------
__global__ __launch_bounds__(256)
void switchgate_norm_kernel(float* __restrict__ out, int SE) {
    int idx = blockIdx.x * blockDim.x + threadIdx.x;
    if (idx >= SE) return;

    float v[8];
    float sum = 0.0f;
#pragma unroll
    for (int b = 0; b < 8; ++b) {
        v[b] = out[(long)b * SE + idx];
        sum += v[b];
    }
    float scale = 8.0f / (sum + 1e-6f);
#pragma unroll
    for (int b = 0; b < 8; ++b)
        out[(long)b * SE + idx] = v[b] * scale;
}

extern "C" void kernel_launch(void* const* d_in, const int* in_sizes, int n_in,
                              void* d_out, int out_size, void* d_ws, size_t ws_size,
                              hipStream_t stream) {
    const float* X    = (const float*)d_in[0];   // [B,S,D]
    const float* W    = (const float*)d_in[1];   // [E,D]
    const float* bias = (const float*)d_in[2];   // [E]
    float* out = (float*)d_out;                  // [B,S,E]

    const int E = in_sizes[2];            // 16
    const int D = in_sizes[1] / E;        // 2048
    const int tokens = in_sizes[0] / D;   // B*S = 32768

    // Kernel 1: one wave per 16-token tile, 8 waves (256 threads) per block.
    const int tiles = tokens / 16;        // 2048, divides exactly
    switchgate_score_kernel<<<tiles / 8, 256, 0, stream>>>(X, W, bias, out, D);

    // Kernel 2: B fixed at 8 (capacity = 8).
    const int SE = out_size / 8;          // S*E = 65536
    switchgate_norm_kernel<<<(SE + 255) / 256, 256, 0, stream>>>(out, SE);
}